// GINEMultiEdgeset_40853728920024
// MI455X (gfx1250) — compile-verified
//
#include <hip/hip_runtime.h>
#include <math.h>

#define Hdim   128
#define Ndim   10000
#define Cdim   4
#define Edim   160000
#define EFdim  16
#define OUTdim 64
#define Ldim   3
#define CNH    (Cdim * Ndim * Hdim)
#define EH     (Edim * Hdim)
#define LSTR   130   // LDS row stride in floats (even -> 8B-aligned b64 loads)

typedef __attribute__((ext_vector_type(2))) float v2f;
typedef __attribute__((ext_vector_type(8))) float v8f;

__device__ __forceinline__ v8f wmma4(v2f a, v2f b, v8f c) {
  // D = A(16x4 f32) * B(4x16 f32) + C(16x16 f32)
  return __builtin_amdgcn_wmma_f32_16x16x4_f32(false, a, false, b, (short)0, c,
                                               false, false);
}

__device__ __forceinline__ float gelu_f(float v) {
  return 0.5f * v * (1.0f + erff(v * 0.7071067811865476f));
}

__device__ __forceinline__ v8f splat8(float v) {
  v8f r;
#pragma unroll
  for (int i = 0; i < 8; ++i) r[i] = v;
  return r;
}

__device__ __forceinline__ void atomicAddF(float* p, float v) {
  (void)__hip_atomic_fetch_add(p, v, __ATOMIC_RELAXED, __HIP_MEMORY_SCOPE_AGENT);
}

// ---------------- zero fill ----------------
__global__ __launch_bounds__(256) void zero_kernel(float* __restrict__ p, int n) {
  int i = blockIdx.x * 256 + threadIdx.x;
  if (i < n) p[i] = 0.0f;
}

// ---------------- edge encoder: emb = edge_attr @ bondW + bondb ----------------
// One wave per 16-edge tile. K=16 (4 k-steps), N=128 (8 n-tiles).
__global__ __launch_bounds__(256) void edge_emb_kernel(
    const float* __restrict__ ea, const float* __restrict__ bw,
    const float* __restrict__ bb, float* __restrict__ emb) {
  const int lane = threadIdx.x & 31;
  const int wid  = blockIdx.x * 8 + (threadIdx.x >> 5);
  const int g = lane >> 4, r = lane & 15;
  const int row0 = wid * 16;
  if (row0 >= Edim) return;  // wave-uniform

  v8f acc[8];
#pragma unroll
  for (int n = 0; n < 8; ++n) acc[n] = splat8(bb[n * 16 + r]);

#pragma unroll
  for (int kk = 0; kk < 4; ++kk) {
    const int kb = kk * 4 + g * 2;
    const float2 av = *(const float2*)(ea + (row0 + r) * EFdim + kb);
    v2f a; a.x = av.x; a.y = av.y;
#pragma unroll
    for (int n = 0; n < 8; ++n) {
      v2f b;
      b.x = bw[kb * Hdim + n * 16 + r];
      b.y = bw[(kb + 1) * Hdim + n * 16 + r];
      acc[n] = wmma4(a, b, acc[n]);
    }
  }
#pragma unroll
  for (int n = 0; n < 8; ++n)
#pragma unroll
    for (int v = 0; v < 8; ++v)
      emb[(row0 + v + g * 8) * Hdim + n * 16 + r] = acc[n][v];
}

// ---------------- message + scatter: agg[dst] += gelu(x[src]+emb)*mask ----------------
// One wave per (edge, channel); lane handles 4 contiguous floats.
__global__ __launch_bounds__(256) void msg_kernel(
    const float* __restrict__ x, const float* __restrict__ emb,
    const float* __restrict__ mask, const int* __restrict__ srcI,
    const int* __restrict__ dstI, float* __restrict__ agg) {
  const int lane = threadIdx.x & 31;
  const int wi = blockIdx.x * 8 + (threadIdx.x >> 5);  // 0 .. E*C-1
  const int e = wi >> 2;
  const int c = wi & 3;
  const int s = srcI[e], d = dstI[e];
  const float m = mask[c * Ndim + s] * mask[c * Ndim + d];
  const float4 xv = *(const float4*)(x + (size_t)(c * Ndim + s) * Hdim + lane * 4);
  const float4 ev = *(const float4*)(emb + (size_t)e * Hdim + lane * 4);
  float* ap = agg + (size_t)(c * Ndim + d) * Hdim + lane * 4;
  atomicAddF(ap + 0, gelu_f(xv.x + ev.x) * m);
  atomicAddF(ap + 1, gelu_f(xv.y + ev.y) * m);
  atomicAddF(ap + 2, gelu_f(xv.z + ev.z) * m);
  atomicAddF(ap + 3, gelu_f(xv.w + ev.w) * m);
}

// ---------------- fused conv layer ----------------
// h = (1+eps)*x + agg ; t = gelu(h@W1+b1) ; u = t@W2+b2 ; xout = x + gelu(u)
// One wave per 16-row tile, 4 waves/block. GEMM1 -> LDS transpose -> GEMM2.
__global__ __launch_bounds__(128) void conv_kernel(
    const float* xin, const float* __restrict__ agg,
    const float* __restrict__ W1, const float* __restrict__ B1,
    const float* __restrict__ W2, const float* __restrict__ B2,
    const float* __restrict__ epsArr, int layer, float* xout) {
  __shared__ float lds[4][16 * LSTR];
  const int lane = threadIdx.x & 31;
  const int w = threadIdx.x >> 5;
  const int g = lane >> 4, r = lane & 15;
  const int row0 = (blockIdx.x * 4 + w) * 16;
  const float ope = 1.0f + epsArr[layer];
  float* L = lds[w];

  v8f acc[8];
#pragma unroll
  for (int n = 0; n < 8; ++n) acc[n] = splat8(B1[n * 16 + r]);
  const float* xrow = xin + (row0 + r) * Hdim;
  const float* arow = agg + (row0 + r) * Hdim;
#pragma unroll 4
  for (int kk = 0; kk < 32; ++kk) {
    const int kb = kk * 4 + g * 2;
    const float2 xv = *(const float2*)(xrow + kb);
    const float2 av = *(const float2*)(arow + kb);
    v2f a; a.x = ope * xv.x + av.x; a.y = ope * xv.y + av.y;
#pragma unroll
    for (int n = 0; n < 8; ++n) {
      v2f b;
      b.x = W1[kb * Hdim + n * 16 + r];
      b.y = W1[(kb + 1) * Hdim + n * 16 + r];
      acc[n] = wmma4(a, b, acc[n]);
    }
  }
  // C-layout -> LDS (A-layout source for GEMM2), with GELU
#pragma unroll
  for (int n = 0; n < 8; ++n)
#pragma unroll
    for (int v = 0; v < 8; ++v)
      L[(v + g * 8) * LSTR + n * 16 + r] = gelu_f(acc[n][v]);
  __syncthreads();

  v8f acc2[8];
#pragma unroll
  for (int n = 0; n < 8; ++n) acc2[n] = splat8(B2[n * 16 + r]);
#pragma unroll 4
  for (int kk = 0; kk < 32; ++kk) {
    const int kb = kk * 4 + g * 2;
    const float2 tv = *(const float2*)(&L[r * LSTR + kb]);
    v2f a; a.x = tv.x; a.y = tv.y;
#pragma unroll
    for (int n = 0; n < 8; ++n) {
      v2f b;
      b.x = W2[kb * Hdim + n * 16 + r];
      b.y = W2[(kb + 1) * Hdim + n * 16 + r];
      acc2[n] = wmma4(a, b, acc2[n]);
    }
  }
#pragma unroll
  for (int n = 0; n < 8; ++n)
#pragma unroll
    for (int v = 0; v < 8; ++v) {
      const int row = row0 + v + g * 8;
      const int col = n * 16 + r;
      const float xo = xin[row * Hdim + col];
      xout[row * Hdim + col] = xo + gelu_f(acc2[n][v]);
    }
}

// ---------------- fused final MLP + masked pooling partial sums ----------------
// y = relu(x@W1+b1)@W2+b2 ; out_acc[c,:] += sum over tile rows of y*mask
__global__ __launch_bounds__(128) void mlp_kernel(
    const float* __restrict__ xin, const float* __restrict__ W1,
    const float* __restrict__ B1, const float* __restrict__ W2,
    const float* __restrict__ B2, const float* __restrict__ mask,
    float* __restrict__ out_acc) {
  __shared__ float lds[4][16 * LSTR];
  const int lane = threadIdx.x & 31;
  const int w = threadIdx.x >> 5;
  const int g = lane >> 4, r = lane & 15;
  const int row0 = (blockIdx.x * 4 + w) * 16;
  float* L = lds[w];

  v8f acc[8];
#pragma unroll
  for (int n = 0; n < 8; ++n) acc[n] = splat8(B1[n * 16 + r]);
  const float* xrow = xin + (row0 + r) * Hdim;
#pragma unroll 4
  for (int kk = 0; kk < 32; ++kk) {
    const int kb = kk * 4 + g * 2;
    const float2 xv = *(const float2*)(xrow + kb);
    v2f a; a.x = xv.x; a.y = xv.y;
#pragma unroll
    for (int n = 0; n < 8; ++n) {
      v2f b;
      b.x = W1[kb * Hdim + n * 16 + r];
      b.y = W1[(kb + 1) * Hdim + n * 16 + r];
      acc[n] = wmma4(a, b, acc[n]);
    }
  }
#pragma unroll
  for (int n = 0; n < 8; ++n)
#pragma unroll
    for (int v = 0; v < 8; ++v)
      L[(v + g * 8) * LSTR + n * 16 + r] = fmaxf(acc[n][v], 0.0f);
  __syncthreads();

  v8f acc2[4];  // OUT=64 -> 4 n-tiles
#pragma unroll
  for (int n = 0; n < 4; ++n) acc2[n] = splat8(B2[n * 16 + r]);
#pragma unroll 4
  for (int kk = 0; kk < 32; ++kk) {
    const int kb = kk * 4 + g * 2;
    const float2 tv = *(const float2*)(&L[r * LSTR + kb]);
    v2f a; a.x = tv.x; a.y = tv.y;
#pragma unroll
    for (int n = 0; n < 4; ++n) {
      v2f b;
      b.x = W2[kb * OUTdim + n * 16 + r];
      b.y = W2[(kb + 1) * OUTdim + n * 16 + r];
      acc2[n] = wmma4(a, b, acc2[n]);
    }
  }
  // masked partial reduction over the 16 rows of this tile
  float mrow[8];
#pragma unroll
  for (int v = 0; v < 8; ++v) mrow[v] = mask[row0 + v + g * 8];
  const int c = row0 / Ndim;  // tiles never straddle channels (10000 % 16 == 0)
#pragma unroll
  for (int n = 0; n < 4; ++n) {
    float s = 0.0f;
#pragma unroll
    for (int v = 0; v < 8; ++v) s += acc2[n][v] * mrow[v];
    s += __shfl_xor(s, 16, 32);  // combine the two row-halves
    if (g == 0) atomicAddF(&out_acc[c * OUTdim + n * 16 + r], s);
  }
}

// ---------------- mask denominator per channel ----------------
__global__ __launch_bounds__(256) void denom_kernel(const float* __restrict__ mask,
                                                    float* __restrict__ denom) {
  const int c = blockIdx.x;
  float s = 0.0f;
  for (int i = threadIdx.x; i < Ndim; i += 256) s += mask[c * Ndim + i];
#pragma unroll
  for (int o = 16; o > 0; o >>= 1) s += __shfl_xor(s, o, 32);
  if ((threadIdx.x & 31) == 0) atomicAddF(&denom[c], s);
}

// ---------------- finalize ----------------
__global__ __launch_bounds__(256) void finalize_kernel(const float* __restrict__ acc,
                                                       const float* __restrict__ denom,
                                                       float* __restrict__ out) {
  const int i = threadIdx.x;  // C*OUT == 256
  out[i] = acc[i] / denom[i >> 6];
}

extern "C" void kernel_launch(void* const* d_in, const int* in_sizes, int n_in,
                              void* d_out, int out_size, void* d_ws, size_t ws_size,
                              hipStream_t stream) {
  (void)in_sizes; (void)n_in; (void)out_size; (void)ws_size;
  const float* x      = (const float*)d_in[0];
  const float* mask   = (const float*)d_in[1];
  const float* ea     = (const float*)d_in[2];
  const float* bondW  = (const float*)d_in[3];
  const float* bondb  = (const float*)d_in[4];
  const float* convW1 = (const float*)d_in[5];
  const float* convb1 = (const float*)d_in[6];
  const float* convW2 = (const float*)d_in[7];
  const float* convb2 = (const float*)d_in[8];
  const float* eps    = (const float*)d_in[9];
  const float* mlpW1  = (const float*)d_in[10];
  const float* mlpb1  = (const float*)d_in[11];
  const float* mlpW2  = (const float*)d_in[12];
  const float* mlpb2  = (const float*)d_in[13];
  const int* eidx     = (const int*)d_in[14];
  const int* srcI = eidx;
  const int* dstI = eidx + Edim;

  char* ws = (char*)d_ws;
  size_t off = 0;
  auto alloc = [&](size_t bytes) {
    void* p = ws + off;
    off = (off + bytes + 255) & ~(size_t)255;
    return p;
  };
  float* x_a     = (float*)alloc((size_t)CNH * 4);
  float* x_b     = (float*)alloc((size_t)CNH * 4);
  float* agg     = (float*)alloc((size_t)CNH * 4);
  float* emb     = (float*)alloc((size_t)EH * 4);
  float* out_acc = (float*)alloc((size_t)(Cdim * OUTdim + Cdim) * 4);
  float* denom   = out_acc + Cdim * OUTdim;

  hipMemcpyAsync(x_a, x, (size_t)CNH * 4, hipMemcpyDeviceToDevice, stream);
  edge_emb_kernel<<<Edim / 16 / 8, 256, 0, stream>>>(ea, bondW, bondb, emb);

  float* cur = x_a;
  float* nxt = x_b;
  const int tiles = Cdim * Ndim / 16;  // 2500
  for (int l = 0; l < Ldim; ++l) {
    zero_kernel<<<(CNH + 255) / 256, 256, 0, stream>>>(agg, CNH);
    msg_kernel<<<(Edim * Cdim) / 8, 256, 0, stream>>>(cur, emb, mask, srcI, dstI, agg);
    conv_kernel<<<tiles / 4, 128, 0, stream>>>(
        cur, agg, convW1 + l * Hdim * Hdim, convb1 + l * Hdim,
        convW2 + l * Hdim * Hdim, convb2 + l * Hdim, eps, l, nxt);
    float* t = cur; cur = nxt; nxt = t;
  }

  zero_kernel<<<1, 256, 0, stream>>>(out_acc, Cdim * OUTdim + Cdim);
  mlp_kernel<<<tiles / 4, 128, 0, stream>>>(cur, mlpW1, mlpb1, mlpW2, mlpb2, mask,
                                            out_acc);
  denom_kernel<<<Cdim, 256, 0, stream>>>(mask, denom);
  finalize_kernel<<<1, Cdim * OUTdim, 0, stream>>>(out_acc, denom, (float*)d_out);
}